// HIDE_11501922419036
// MI455X (gfx1250) — compile-verified
//
#include <hip/hip_runtime.h>
#include <math.h>

#define NN 100      // N == E == 100
#define NP 112      // padded to 7 tiles of 16
#define DD 64       // feature dim, 4 tiles of 16
#define NEGV (-9e15f)

typedef __attribute__((ext_vector_type(2))) float v2f;
typedef __attribute__((ext_vector_type(8))) float v8f;

__device__ __forceinline__ v8f wmma4(v2f a, v2f b, v8f c) {
  // D = A(16x4,f32) x B(4x16,f32) + C(16x16,f32)
  return __builtin_amdgcn_wmma_f32_16x16x4_f32(false, a, false, b, (short)0, c,
                                               false, false);
}

__device__ __forceinline__ float lrelu(float x) { return x >= 0.f ? x : 0.2f * x; }

struct Smem {
  float h[NN * DD];          // 25600 B  current node embeddings
  float edge[NN * DD];       // 25600 B  edge_cluster, then edge (in-place per strip)
  float sbuf[NP * 16];       // 7168 B   attention strip (rows x 16)
  float avec[6 * DD];        // 1536 B   a10,a11,a12,a20,a21,a22
  float sc[DD];              // 256 B
  float red[256];            // 1024 B   softmax reduction scratch
  unsigned int code[NP * 7]; // 3136 B   H codes, 2 bits each, 16 per word
};                           // total 64320 B

// 2-bit H code lookup (0 outside the real N x E region)
#define GETC(nn_, ee_) \
  ((sm.code[(nn_) * 7 + ((ee_) >> 4)] >> (((ee_) & 15) << 1)) & 3u)

// Softmax over the first index of sbuf[row][col], rows 0..NP-1, cols 0..15.
__device__ void softmax_cols(float* sbuf, float* red, int tid) {
  const int col = tid & 15;
  const int grp = tid >> 4; // 16 groups x 16 cols
  float m = -3.0e38f;
  for (int r = grp; r < NP; r += 16) m = fmaxf(m, sbuf[r * 16 + col]);
  red[grp * 16 + col] = m;
  __syncthreads();
  if (grp == 0) {
    float mm = red[col];
#pragma unroll
    for (int g = 1; g < 16; ++g) mm = fmaxf(mm, red[g * 16 + col]);
    red[col] = mm;
  }
  __syncthreads();
  const float colmax = red[col];
  __syncthreads();
  float s = 0.f;
  for (int r = grp; r < NP; r += 16) {
    float v = __expf(sbuf[r * 16 + col] - colmax); // exp(NEG-max) -> 0
    sbuf[r * 16 + col] = v;
    s += v;
  }
  red[grp * 16 + col] = s;
  __syncthreads();
  if (grp == 0) {
    float ss = 0.f;
#pragma unroll
    for (int g = 0; g < 16; ++g) ss += red[g * 16 + col];
    red[col] = 1.0f / ss;
  }
  __syncthreads();
  const float inv = red[col];
  __syncthreads();
  for (int r = grp; r < NP; r += 16) sbuf[r * 16 + col] *= inv;
  __syncthreads();
}

__global__ __launch_bounds__(256) void gat_hyper_kernel(
    const float* __restrict__ hidden, const float* __restrict__ Hm,
    const float* __restrict__ s_c, const float* __restrict__ a10,
    const float* __restrict__ a11, const float* __restrict__ a12,
    const float* __restrict__ a20, const float* __restrict__ a21,
    const float* __restrict__ a22, float* __restrict__ out) {
  __shared__ Smem sm;
  const int tid = threadIdx.x;
  const int b = blockIdx.x;
  const int lane = tid & 31;
  const int wv = tid >> 5;            // 8 waves
  const int lm = lane & 15;           // M (A) / N (B,C) lane index
  const int kb = (lane >> 4) << 1;    // K base for A/B fragments
  const int rb = (lane >> 4) << 3;    // row base for C fragment

  // ---------------- Phase 0: stage inputs ----------------
  for (int i = tid; i < NN * DD; i += 256) sm.h[i] = hidden[b * NN * DD + i];
  if (tid < DD) {
    sm.sc[tid] = s_c[b * DD + tid];
    sm.avec[0 * DD + tid] = a10[tid];
    sm.avec[1 * DD + tid] = a11[tid];
    sm.avec[2 * DD + tid] = a12[tid];
    sm.avec[3 * DD + tid] = a20[tid];
    sm.avec[4 * DD + tid] = a21[tid];
    sm.avec[5 * DD + tid] = a22[tid];
  }
  for (int w = tid; w < NP * 7; w += 256) {
    const int n = w / 7, eg = w % 7;
    unsigned int pk = 0u;
    if (n < NN) {
#pragma unroll
      for (int j = 0; j < 16; ++j) {
        const int e = eg * 16 + j;
        if (e < NN) {
          const int cv = (int)Hm[(b * NN + n) * NN + e]; // exact 0..3
          pk |= ((unsigned)cv & 3u) << (2 * j);
        }
      }
    }
    sm.code[w] = pk;
  }
  __syncthreads();

  for (int layer = 0; layer < 2; ++layer) {
    // ---------- Phase 1: edge_cluster = H_adj^T * h  (E x N * N x D) ----------
    for (int t = wv; t < 28; t += 8) {
      const int te = t >> 2, td = t & 3;
      const int eA = te * 16 + lm;  // A row m = e
      const int dB = td * 16 + lm;  // B col n = d
      v8f c = {};
      for (int kk = 0; kk < 28; ++kk) {
        const int n0 = kk * 4 + kb; // K = node index
        v2f a, bv;
        a.x = (GETC(n0 + 0, eA) != 0u) ? 1.f : 0.f;
        a.y = (GETC(n0 + 1, eA) != 0u) ? 1.f : 0.f;
        bv.x = (n0 + 0 < NN) ? sm.h[(n0 + 0) * DD + dB] : 0.f;
        bv.y = (n0 + 1 < NN) ? sm.h[(n0 + 1) * DD + dB] : 0.f;
        c = wmma4(a, bv, c);
      }
#pragma unroll
      for (int r = 0; r < 8; ++r) {
        const int e = te * 16 + rb + r;
        if (e < NN) sm.edge[e * DD + dB] = c[r];
      }
    }
    __syncthreads();

    // ---------- Phase 2: node->edge attention, per 16-wide e-strip ----------
    for (int s = 0; s < 7; ++s) {
      const int e0 = s * 16;
      // (a) logits: atts_k[n,e] = sum_d h[n,d] * edge_cluster[e,d] * a1k[d]
      for (int t = wv; t < 7; t += 8) {
        const int nA = t * 16 + lm;
        const int eB = e0 + lm;
        v8f c0 = {}, c1 = {}, c2 = {};
        for (int kk = 0; kk < 16; ++kk) {
          const int d0 = kk * 4 + kb;
          const bool okn = nA < NN;
          const bool oke = eB < NN;
          v2f a;
          a.x = okn ? sm.h[nA * DD + d0] : 0.f;
          a.y = okn ? sm.h[nA * DD + d0 + 1] : 0.f;
          const float e0v = oke ? sm.edge[eB * DD + d0] : 0.f;
          const float e1v = oke ? sm.edge[eB * DD + d0 + 1] : 0.f;
          v2f b0, b1, b2;
          b0.x = e0v * sm.avec[0 * DD + d0]; b0.y = e1v * sm.avec[0 * DD + d0 + 1];
          b1.x = e0v * sm.avec[1 * DD + d0]; b1.y = e1v * sm.avec[1 * DD + d0 + 1];
          b2.x = e0v * sm.avec[2 * DD + d0]; b2.y = e1v * sm.avec[2 * DD + d0 + 1];
          c0 = wmma4(a, b0, c0);
          c1 = wmma4(a, b1, c1);
          c2 = wmma4(a, b2, c2);
        }
#pragma unroll
        for (int r = 0; r < 8; ++r) {
          const int n = t * 16 + rb + r;
          const int e = e0 + lm;
          const unsigned cd = GETC(n, e);
          const float l0 = lrelu(c0[r]);
          const float l1 = lrelu(c1[r]);
          const float l2 = lrelu(c2[r]);
          float v = NEGV;
          v = (cd == 1u) ? l0 : v;
          v = (cd == 2u) ? l1 : v;
          v = (cd == 3u) ? l2 : v;
          sm.sbuf[n * 16 + lm] = v;
        }
      }
      __syncthreads();
      // (b) softmax over n (axis=1) per e column
      softmax_cols(sm.sbuf, sm.red, tid);
      // (c) edge[e in strip, d] = sum_n alpha1[n,e] * h[n,d]  (in-place rows)
      for (int t = wv; t < 4; t += 8) {
        const int dB = t * 16 + lm;
        v8f c = {};
        for (int kk = 0; kk < 28; ++kk) {
          const int n0 = kk * 4 + kb;
          v2f a, bv;
          a.x = sm.sbuf[(n0 + 0) * 16 + lm]; // A[m=e_local][k=n]
          a.y = sm.sbuf[(n0 + 1) * 16 + lm];
          bv.x = (n0 + 0 < NN) ? sm.h[(n0 + 0) * DD + dB] : 0.f;
          bv.y = (n0 + 1 < NN) ? sm.h[(n0 + 1) * DD + dB] : 0.f;
          c = wmma4(a, bv, c);
        }
#pragma unroll
        for (int r = 0; r < 8; ++r) {
          const int e = e0 + rb + r;
          if (e < NN) sm.edge[e * DD + dB] = c[r];
        }
      }
      __syncthreads();
    }

    // ---------- Phase 3: edge->node attention, per 16-wide n-strip ----------
    for (int s = 0; s < 7; ++s) {
      const int n0s = s * 16;
      // (a) logits: atts2_k[n,e] = sum_d (h[n,d]+sc[d]) * edge[e,d] * a2k[d]
      for (int t = wv; t < 7; t += 8) {
        const int nA = n0s + lm;
        const int eB = t * 16 + lm;
        v8f c0 = {}, c1 = {}, c2 = {};
        for (int kk = 0; kk < 16; ++kk) {
          const int d0 = kk * 4 + kb;
          const bool okn = nA < NN;
          const bool oke = eB < NN;
          v2f a;
          a.x = okn ? (sm.h[nA * DD + d0] + sm.sc[d0]) : 0.f;
          a.y = okn ? (sm.h[nA * DD + d0 + 1] + sm.sc[d0 + 1]) : 0.f;
          const float e0v = oke ? sm.edge[eB * DD + d0] : 0.f;
          const float e1v = oke ? sm.edge[eB * DD + d0 + 1] : 0.f;
          v2f b0, b1, b2;
          b0.x = e0v * sm.avec[3 * DD + d0]; b0.y = e1v * sm.avec[3 * DD + d0 + 1];
          b1.x = e0v * sm.avec[4 * DD + d0]; b1.y = e1v * sm.avec[4 * DD + d0 + 1];
          b2.x = e0v * sm.avec[5 * DD + d0]; b2.y = e1v * sm.avec[5 * DD + d0 + 1];
          c0 = wmma4(a, b0, c0);
          c1 = wmma4(a, b1, c1);
          c2 = wmma4(a, b2, c2);
        }
#pragma unroll
        for (int r = 0; r < 8; ++r) {
          const int m = rb + r;
          const int n = n0s + m;
          const int e = t * 16 + lm;
          const unsigned cd = GETC(n, e);
          const float l0 = lrelu(c0[r]);
          const float l1 = lrelu(c1[r]);
          const float l2 = lrelu(c2[r]);
          float v = NEGV;
          v = (cd == 1u) ? l0 : v;
          v = (cd == 2u) ? l1 : v;
          v = (cd == 3u) ? l2 : v;
          sm.sbuf[e * 16 + m] = v; // alpha2 strip: [e][n_local]
        }
      }
      __syncthreads();
      // (b) softmax over e (axis=2) per n row
      softmax_cols(sm.sbuf, sm.red, tid);
      // (c) h_new[n in strip, d] = sum_e alpha2[n,e] * edge[e,d]
      for (int t = wv; t < 4; t += 8) {
        const int dB = t * 16 + lm;
        v8f c = {};
        for (int kk = 0; kk < 28; ++kk) {
          const int ek = kk * 4 + kb;
          v2f a, bv;
          a.x = sm.sbuf[(ek + 0) * 16 + lm]; // A[m=n_local][k=e]
          a.y = sm.sbuf[(ek + 1) * 16 + lm];
          bv.x = (ek + 0 < NN) ? sm.edge[(ek + 0) * DD + dB] : 0.f;
          bv.y = (ek + 1 < NN) ? sm.edge[(ek + 1) * DD + dB] : 0.f;
          c = wmma4(a, bv, c);
        }
#pragma unroll
        for (int r = 0; r < 8; ++r) {
          const int n = n0s + rb + r;
          if (n < NN) {
            const float v = c[r];
            sm.h[n * DD + dB] = v; // becomes h_emb for next layer
            const int gi = (b * NN + n) * DD + dB;
            if (layer == 0) out[gi] = v;   // overwrite poison
            else out[gi] += v;             // h_out accumulation
          }
        }
      }
      __syncthreads();
    }
  }
}

extern "C" void kernel_launch(void* const* d_in, const int* in_sizes, int n_in,
                              void* d_out, int out_size, void* d_ws,
                              size_t ws_size, hipStream_t stream) {
  const float* hidden = (const float*)d_in[0];
  const float* Hm = (const float*)d_in[1];
  const float* s_c = (const float*)d_in[2];
  const float* a10 = (const float*)d_in[3];
  const float* a11 = (const float*)d_in[4];
  const float* a12 = (const float*)d_in[5];
  const float* a20 = (const float*)d_in[6];
  const float* a21 = (const float*)d_in[7];
  const float* a22 = (const float*)d_in[8];
  float* out = (float*)d_out;
  const int Bn = in_sizes[0] / (NN * DD); // 128 batches -> 128 workgroups
  gat_hyper_kernel<<<dim3(Bn), dim3(256), 0, stream>>>(
      hidden, Hm, s_c, a10, a11, a12, a20, a21, a22, out);
  (void)n_in; (void)out_size; (void)d_ws; (void)ws_size;
}